// CircularMemoryBank_26319559590179
// MI455X (gfx1250) — compile-verified
//
#include <hip/hip_runtime.h>
#include <math.h>

typedef __attribute__((ext_vector_type(2))) float v2f;
typedef __attribute__((ext_vector_type(8))) float v8f;

#define VEC_DIM   4096
#define WGS       128            // 4 waves of 32 (wave32)
#define NWG_STORE 256
#define PAIRS_PER_WG (4096 / NWG_STORE)

// Constant DFT-16 matrix, striped per the 16x4 f32 A-operand layout:
// lanes 0-15 hold M=0..15; VGPR0/1 = K pair; lanes 16-31 = K+2 pair.
struct F16Mat { v2f ar[4]; v2f ai[4]; v2f nai[4]; };

__device__ __forceinline__ F16Mat make_dft16(int lane) {
  F16Mat F;
  const int m   = lane & 15;
  const int kof = ((lane >> 4) & 1) << 1;
  const float w = -6.28318530717958647692f / 16.0f;   // exp(-2*pi*i*m*k/16)
  #pragma unroll
  for (int kk = 0; kk < 4; ++kk) {
    const int k0 = kk * 4 + kof;
    const int k1 = k0 + 1;
    const float a0 = w * (float)((m * k0) & 15);
    const float a1 = w * (float)((m * k1) & 15);
    const float c0 = __cosf(a0), s0 = __sinf(a0);
    const float c1 = __cosf(a1), s1 = __sinf(a1);
    F.ar[kk]  = v2f{ c0,  c1};
    F.ai[kk]  = v2f{ s0,  s1};
    F.nai[kk] = v2f{-s0, -s1};
  }
  return F;
}

// One complex 16x16x16 matmul: (Fr + i*Fi) @ (Xr + i*Xi), K split in 4 chunks
// of 4 for v_wmma_f32_16x16x4_f32.  B operand gathered from LDS at
// base + row*rowStride + col*colStride (row = contraction index).
__device__ __forceinline__ void dft16_tile(const F16Mat& F, const float2* src,
                                           int base, int rowStride, int colStride,
                                           int lane, v8f& cr, v8f& ci) {
  const int n   = lane & 15;
  const int kof = ((lane >> 4) & 1) << 1;
  v2f br[4], bi[4];
  #pragma unroll
  for (int kk = 0; kk < 4; ++kk) {
    const int r0 = kk * 4 + kof;
    const float2 e0 = src[base + (r0    ) * rowStride + n * colStride];
    const float2 e1 = src[base + (r0 + 1) * rowStride + n * colStride];
    br[kk] = v2f{e0.x, e1.x};
    bi[kk] = v2f{e0.y, e1.y};
  }
  cr = v8f{0.f,0.f,0.f,0.f,0.f,0.f,0.f,0.f};
  ci = v8f{0.f,0.f,0.f,0.f,0.f,0.f,0.f,0.f};
  #pragma unroll
  for (int kk = 0; kk < 4; ++kk)
    cr = __builtin_amdgcn_wmma_f32_16x16x4_f32(false, F.ar[kk],  false, br[kk], (short)0, cr, false, false);
  #pragma unroll
  for (int kk = 0; kk < 4; ++kk)
    cr = __builtin_amdgcn_wmma_f32_16x16x4_f32(false, F.nai[kk], false, bi[kk], (short)0, cr, false, false);
  #pragma unroll
  for (int kk = 0; kk < 4; ++kk)
    ci = __builtin_amdgcn_wmma_f32_16x16x4_f32(false, F.ar[kk],  false, bi[kk], (short)0, ci, false, false);
  #pragma unroll
  for (int kk = 0; kk < 4; ++kk)
    ci = __builtin_amdgcn_wmma_f32_16x16x4_f32(false, F.ai[kk],  false, br[kk], (short)0, ci, false, false);
}

// 4096-pt complex FFT, radix-16^3 via WMMA; input in ping, natural-order
// result in pong.  n = 256*n2 + 16*n1 + n0 ; k = k0 + 16*k1 + 256*k2.
__device__ __forceinline__ void fft4096(float2* ping, float2* pong,
                                        const F16Mat& F, int tid) {
  const int lane  = tid & 31;
  const int wave  = tid >> 5;
  const int n     = lane & 15;
  const int mbase = ((lane >> 4) & 1) << 3;
  __syncthreads();
  // ---- stage 1: contract n2 ;  out[k0][n1][n0] *= W256^(n1*k0) ----
  for (int t = 0; t < 4; ++t) {
    const int n1 = wave * 4 + t;
    v8f cr, ci;
    dft16_tile(F, ping, 16 * n1, 256, 1, lane, cr, ci);
    #pragma unroll
    for (int j = 0; j < 8; ++j) {
      const int k0 = mbase + j;
      const float ang = (-6.28318530717958647692f / 256.0f) * (float)((n1 * k0) & 255);
      const float c = __cosf(ang), s = __sinf(ang);
      const float xr = cr[j], xi = ci[j];
      pong[256 * k0 + 16 * n1 + n] = make_float2(xr * c - xi * s, xr * s + xi * c);
    }
  }
  __syncthreads();
  // ---- stage 2: contract n1 ;  out[k0][k1][n0] *= W4096^(n0*(k0+16*k1)) ----
  for (int t = 0; t < 4; ++t) {
    const int k0 = wave * 4 + t;
    v8f cr, ci;
    dft16_tile(F, pong, 256 * k0, 16, 1, lane, cr, ci);
    #pragma unroll
    for (int j = 0; j < 8; ++j) {
      const int k1 = mbase + j;
      const float ang = (-6.28318530717958647692f / 4096.0f) * (float)((n * (k0 + 16 * k1)) & 4095);
      const float c = __cosf(ang), s = __sinf(ang);
      const float xr = cr[j], xi = ci[j];
      ping[256 * k0 + 16 * k1 + n] = make_float2(xr * c - xi * s, xr * s + xi * c);
    }
  }
  __syncthreads();
  // ---- stage 3: contract n0 ; digit-reversed natural-order store ----
  for (int t = 0; t < 4; ++t) {
    const int k0 = wave * 4 + t;
    v8f cr, ci;
    dft16_tile(F, ping, 256 * k0, 1, 16, lane, cr, ci);
    const int k1 = n;
    #pragma unroll
    for (int j = 0; j < 8; ++j) {
      const int k2 = mbase + j;
      pong[k0 + 16 * k1 + 256 * k2] = make_float2(cr[j], ci[j]);
    }
  }
  __syncthreads();
}

// ---- Pass 1: one packed FFT per pair.  z = k + i*v, Z = FK + i*FV, and
//      FK[j]*FV[j] = -i*(Z[j]^2 - conj(Z[N-j]^2))/4  (k, v real). ----
__global__ __launch_bounds__(WGS) void hrr_store(const float* __restrict__ K,
                                                 const float* __restrict__ V,
                                                 float2* __restrict__ partial) {
  __shared__ float2 ping[VEC_DIM];
  __shared__ float2 pong[VEC_DIM];
  const int tid = threadIdx.x;
  const F16Mat F = make_dft16(tid & 31);
  float2 acc[32];
  #pragma unroll
  for (int j = 0; j < 32; ++j) acc[j] = make_float2(0.f, 0.f);
  const int pair0 = blockIdx.x * PAIRS_PER_WG;
  for (int p = 0; p < PAIRS_PER_WG; ++p) {
    const float* krow = K + (size_t)(pair0 + p) * VEC_DIM;
    const float* vrow = V + (size_t)(pair0 + p) * VEC_DIM;
    __builtin_prefetch(krow + VEC_DIM + tid, 0, 1);
    __builtin_prefetch(vrow + VEC_DIM + tid, 0, 1);
    #pragma unroll
    for (int j = 0; j < 32; ++j) {
      const int idx = tid + WGS * j;
      ping[idx] = make_float2(krow[idx], vrow[idx]);   // z = k + i*v
    }
    fft4096(ping, pong, F, tid);                       // pong = Z
    #pragma unroll
    for (int j = 0; j < 32; ++j) {
      const int idx = tid + WGS * j;
      const float2 Zj = pong[idx];
      const float2 Zm = pong[(VEC_DIM - idx) & (VEC_DIM - 1)];
      // A2 = Z[j]^2 ; B2 = conj(Z[N-j]^2)
      const float2 A2 = make_float2(Zj.x * Zj.x - Zj.y * Zj.y,  2.f * Zj.x * Zj.y);
      const float2 B2 = make_float2(Zm.x * Zm.x - Zm.y * Zm.y, -2.f * Zm.x * Zm.y);
      const float dx = A2.x - B2.x;
      const float dy = A2.y - B2.y;
      acc[j].x += 0.25f * dy;     // P = -i*(A2-B2)/4
      acc[j].y -= 0.25f * dx;
    }
  }
  float2* dst = partial + (size_t)blockIdx.x * VEC_DIM;
  #pragma unroll
  for (int j = 0; j < 32; ++j) dst[tid + WGS * j] = acc[j];
}

// ---- Pass 2: FM[k] = sum over workgroup partials (deterministic) ----
__global__ __launch_bounds__(128) void hrr_reduce(const float2* __restrict__ partial,
                                                  float2* __restrict__ fm) {
  const int k = blockIdx.x * blockDim.x + threadIdx.x;
  if (k >= VEC_DIM) return;
  float sr = 0.f, si = 0.f;
  for (int w = 0; w < NWG_STORE; ++w) {
    const float2 v = partial[(size_t)w * VEC_DIM + k];
    sr += v.x; si += v.y;
  }
  fm[k] = make_float2(sr, si);
}

// ---- Pass 3: two queries per WG.  z = q0 + i*q1 -> Z = FQ0 + i*FQ1.
//      H0+iH1 = conj(FM)*Z/N ; FFT(H0+iH1) = out0 + i*out1 (both real). ----
__global__ __launch_bounds__(WGS) void hrr_retrieve(const float* __restrict__ Q,
                                                    const float2* __restrict__ fm,
                                                    float* __restrict__ out) {
  __shared__ float2 ping[VEC_DIM];
  __shared__ float2 pong[VEC_DIM];
  const int tid = threadIdx.x;
  const F16Mat F = make_dft16(tid & 31);
  const size_t q0 = (size_t)blockIdx.x * 2;
  const float* q0row = Q + q0 * VEC_DIM;
  const float* q1row = Q + (q0 + 1) * VEC_DIM;
  #pragma unroll
  for (int j = 0; j < 32; ++j) {
    const int idx = tid + WGS * j;
    ping[idx] = make_float2(q0row[idx], q1row[idx]);   // z = q0 + i*q1
  }
  fft4096(ping, pong, F, tid);                         // pong = Z
  const float inv = 1.0f / (float)VEC_DIM;
  #pragma unroll
  for (int j = 0; j < 32; ++j) {
    const int idx = tid + WGS * j;
    const float2 m = fm[idx];
    const float2 z = pong[idx];
    ping[idx] = make_float2((m.x * z.x + m.y * z.y) * inv,    // conj(FM)*Z / N
                            (m.x * z.y - m.y * z.x) * inv);
  }
  fft4096(ping, pong, F, tid);
  float* o0 = out + q0 * VEC_DIM;
  float* o1 = out + (q0 + 1) * VEC_DIM;
  #pragma unroll
  for (int j = 0; j < 32; ++j) {
    const int idx = tid + WGS * j;
    const float2 w = pong[idx];
    o0[idx] = w.x;
    o1[idx] = w.y;
  }
}

extern "C" void kernel_launch(void* const* d_in, const int* in_sizes, int n_in,
                              void* d_out, int out_size, void* d_ws, size_t ws_size,
                              hipStream_t stream) {
  (void)in_sizes; (void)n_in; (void)out_size; (void)ws_size;
  const float* keys    = (const float*)d_in[0];
  const float* values  = (const float*)d_in[1];
  const float* queries = (const float*)d_in[2];
  float* out = (float*)d_out;

  float2* partial = (float2*)d_ws;                          // NWG_STORE * 4096 float2 (8 MB)
  float2* fm      = partial + (size_t)NWG_STORE * VEC_DIM;  // 4096 float2 (32 KB)

  hipLaunchKernelGGL(hrr_store,    dim3(NWG_STORE),   dim3(WGS), 0, stream, keys, values, partial);
  hipLaunchKernelGGL(hrr_reduce,   dim3(VEC_DIM/128), dim3(128), 0, stream, partial, fm);
  hipLaunchKernelGGL(hrr_retrieve, dim3(VEC_DIM/2),   dim3(WGS), 0, stream, queries, fm, out);
}